// RhoLoss_57037165691054
// MI455X (gfx1250) — compile-verified
//
#include <hip/hip_runtime.h>
#include <hip/hip_bf16.h>

typedef __attribute__((ext_vector_type(2))) float v2f;
typedef __attribute__((ext_vector_type(4))) float v4f;
typedef __attribute__((ext_vector_type(8))) float v8f;

// ---------------------------------------------------------------------------
// Zero the scalar output (harness poisons d_out; graph replays must re-zero).
// ---------------------------------------------------------------------------
__global__ void rho_zero_out(float* out) {
    if (threadIdx.x == 0) out[0] = 0.0f;
}

// ---------------------------------------------------------------------------
// delta = c_in - c_tgt, vectorized b128 (element counts are multiples of 1024).
// ---------------------------------------------------------------------------
__global__ void rho_delta4(const v4f* __restrict__ a,
                           const v4f* __restrict__ b,
                           v4f* __restrict__ d, int n4) {
    int i = blockIdx.x * 256 + threadIdx.x;
    if (i < n4) d[i] = a[i] - b[i];
}

// ---------------------------------------------------------------------------
// Wave32 sum via V_WMMA_F32_16X16X4_F32 with an all-ones B:
// D[m,n] = sum_k A[m,k] (column-independent). Each lane's partial occupies one
// A element (a1=0), so summing the 8 D VGPRs gives rows 0..7 (lanes 0..15) /
// rows 8..15 (lanes 16..31); one SWAPX16 ds_swizzle + add yields the full
// 32-lane sum in every lane.
// ---------------------------------------------------------------------------
__device__ __forceinline__ float wave_sum_wmma(float x) {
    v2f a; a.x = x;    a.y = 0.0f;
    v2f b; b.x = 1.0f; b.y = 1.0f;
    v8f c = {};
    c = __builtin_amdgcn_wmma_f32_16x16x4_f32(
        /*neg_a=*/false, a, /*neg_b=*/false, b,
        /*c_mod=*/(short)0, c, /*reuse_a=*/false, /*reuse_b=*/false);
    float p = ((c[0] + c[1]) + (c[2] + c[3])) + ((c[4] + c[5]) + (c[6] + c[7]));
    // lane ^= 16 exchange: ds_swizzle group-of-32, and=0x1F | or=0 | xor=0x10
    int q = __builtin_amdgcn_ds_swizzle(__float_as_int(p), 0x401F);
    return p + __int_as_float(q);
}

// ---------------------------------------------------------------------------
// One pair (l1,l2): factor * sum_{s,r,c} d1[s,r]*O[s,r,c]*d2[s,c].
// grid.x = K1 blocks, 256 threads = 8 wave32s, one wave per (s,row).
// K1,K2 compile-time -> inner loop (K2/128 iters, 2..14) fully unrolls so all
// NT b128 loads of the one-shot O stream issue before any wait (max MLP).
// d2[s] (<= 7 KB) is LDS-resident; accumulate in v4f -> v_pk_fma_f32.
// K2 % 128 == 0 -> uniform trip counts, EXEC all-ones at the WMMA.
// ---------------------------------------------------------------------------
template <int K1, int K2>
__global__ __launch_bounds__(256)
void rho_pair(const float* __restrict__ O,
              const float* __restrict__ d1,
              const float* __restrict__ d2,
              float factor, float* __restrict__ out) {
    __shared__ float sd2[K2];
    __shared__ float swsum[8];

    const int rowBase = blockIdx.x * 8;     // K1 % 8 == 0 -> one structure/block
    const int s       = rowBase / K1;       // compile-time strength-reduced

    // Preload d2[s] into LDS with b128 accesses (reused by all 8 rows).
    {
        const v4f* __restrict__ d2s = (const v4f*)(d2 + (size_t)s * K2);
        v4f* sd2v = (v4f*)sd2;
#pragma unroll
        for (int i = threadIdx.x; i < K2 / 4; i += 256) sd2v[i] = d2s[i];
    }
    __syncthreads();

    const int wave   = threadIdx.x >> 5;
    const int lane   = threadIdx.x & 31;
    const int rowInS = (rowBase % K1) + wave;

    const float d1v = d1[(size_t)s * K1 + rowInS];
    const float* __restrict__ Orow = O + ((size_t)s * K1 + rowInS) * (size_t)K2;

    v4f accv = {0.0f, 0.0f, 0.0f, 0.0f};
#pragma unroll
    for (int it = 0; it < K2 / 128; ++it) {
        const int c = it * 128 + lane * 4;
        v4f o = __builtin_nontemporal_load((const v4f*)(Orow + c)); // NT stream
        v4f t = *(const v4f*)(&sd2[c]);                             // ds_load_b128
        accv += o * t;                                              // pk_fma x2
    }
    float acc = ((accv.x + accv.y) + (accv.z + accv.w)) * d1v;

    float wsum = wave_sum_wmma(acc);
    if (lane == 0) swsum[wave] = wsum;
    __syncthreads();

    if (threadIdx.x == 0) {
        float bs = 0.0f;
#pragma unroll
        for (int w = 0; w < 8; ++w) bs += swsum[w];
        atomicAdd(out, factor * bs);
    }
}

// ---------------------------------------------------------------------------
// Launch: zero out -> 4 delta kernels -> 10 templated pair kernels.
// Input order: c_in_l0,c_tgt_l0,...,c_in_l3,c_tgt_l3, then ovlp pairs
// (0,0),(0,1),(0,2),(0,3),(1,1),(1,2),(1,3),(2,2),(2,3),(3,3).
// ---------------------------------------------------------------------------
extern "C" void kernel_launch(void* const* d_in, const int* in_sizes, int n_in,
                              void* d_out, int out_size, void* d_ws, size_t ws_size,
                              hipStream_t stream) {
    (void)in_sizes; (void)n_in; (void)out_size; (void)ws_size;

    static const int Kl[4] = {256, 768, 1280, 1792};   // 32*(2l+1)*8
    const int S = 8;

    float* dw = (float*)d_ws;                          // 128 KB of deltas
    size_t off[4];
    off[0] = 0;
    for (int l = 1; l < 4; ++l) off[l] = off[l - 1] + (size_t)S * Kl[l - 1];

    rho_zero_out<<<1, 32, 0, stream>>>((float*)d_out);

    for (int l = 0; l < 4; ++l) {
        int n4 = (S * Kl[l]) / 4;
        rho_delta4<<<(n4 + 255) / 256, 256, 0, stream>>>(
            (const v4f*)d_in[2 * l], (const v4f*)d_in[2 * l + 1],
            (v4f*)(dw + off[l]), n4);
    }

#define LAUNCH_PAIR(P, L1, L2)                                                  \
    rho_pair<(32 * (2 * (L1) + 1) * 8), (32 * (2 * (L2) + 1) * 8)>              \
        <<<(32 * (2 * (L1) + 1) * 8), 256, 0, stream>>>(                        \
            (const float*)d_in[8 + (P)], dw + off[(L1)], dw + off[(L2)],        \
            ((L1) == (L2)) ? 1.0f : 2.0f, (float*)d_out)

    LAUNCH_PAIR(0, 0, 0);
    LAUNCH_PAIR(1, 0, 1);
    LAUNCH_PAIR(2, 0, 2);
    LAUNCH_PAIR(3, 0, 3);
    LAUNCH_PAIR(4, 1, 1);
    LAUNCH_PAIR(5, 1, 2);
    LAUNCH_PAIR(6, 1, 3);
    LAUNCH_PAIR(7, 2, 2);
    LAUNCH_PAIR(8, 2, 3);
    LAUNCH_PAIR(9, 3, 3);
#undef LAUNCH_PAIR
}